// GDNN_26027501813798
// MI455X (gfx1250) — compile-verified
//
#include <hip/hip_runtime.h>
#include <math.h>

#define NN 50000
#define EE 800000
#define CC 48
#define C3 144
#define C9 432
#define TT 128
#define BN_EPS 1e-5f

typedef float v2f __attribute__((ext_vector_type(2)));
typedef float v8f __attribute__((ext_vector_type(8)));

// Native f32 atomic add (global_atomic_add_f32), not a CAS loop.
__device__ __forceinline__ void atomAddF(float* p, float v) {
    unsafeAtomicAdd(p, v);
}

// ---------------- time embedding MLP (tiny, one block) ----------------
__global__ void k_time(const float* __restrict__ ts,
                       const float* __restrict__ Wtm, const float* __restrict__ btm,
                       const float* __restrict__ Wt1, const float* __restrict__ bt1,
                       const float* __restrict__ Wt2, const float* __restrict__ bt2,
                       float* __restrict__ te1, float* __restrict__ te2) {
    __shared__ float s_emb[TT];
    __shared__ float s_t[TT];
    int tid = threadIdx.x;
    float tval = ts[0];
    if (tid < TT / 2) {
        float f = expf((float)tid * (-logf(10000.f) / (TT / 2 - 1)));
        float a = tval * f;
        s_emb[tid] = sinf(a);
        s_emb[tid + TT / 2] = cosf(a);
    }
    __syncthreads();
    if (tid < TT) {
        float acc = btm[tid];
        for (int j = 0; j < TT; ++j) acc += s_emb[j] * Wtm[j * TT + tid];
        s_t[tid] = fmaxf(acc, 0.f);
    }
    __syncthreads();
    for (int c = tid; c < C9; c += blockDim.x) {
        float acc = bt1[c];
        for (int k = 0; k < TT; ++k) acc += s_t[k] * Wt1[k * C9 + c];
        te1[c] = fmaxf(acc, 0.f);
    }
    for (int c = tid; c < C3; c += blockDim.x) {
        float acc = bt2[c];
        for (int k = 0; k < TT; ++k) acc += s_t[k] * Wt2[k * C3 + c];
        te2[c] = fmaxf(acc, 0.f);
    }
}

// ---------------- helpers ----------------
__global__ void k_fill(float* __restrict__ p, int n, float v) {
    int i = blockIdx.x * blockDim.x + threadIdx.x;
    if (i < n) p[i] = v;
}

__global__ void k_degcnt(const int* __restrict__ dst, float* __restrict__ deg, int e) {
    int i = blockIdx.x * blockDim.x + threadIdx.x;
    if (i < e) atomAddF(&deg[dst[i]], 1.f);
}

__global__ void k_dinv(const float* __restrict__ deg, float* __restrict__ dinv,
                       float* __restrict__ dinv2, int n) {
    int i = blockIdx.x * blockDim.x + threadIdx.x;
    if (i >= n) return;
    float d = deg[i];
    dinv[i] = rsqrtf(d);
    dinv2[i] = 1.f / d;
}

__global__ void k_enorm(const int* __restrict__ src, const int* __restrict__ dst,
                        const float* __restrict__ dinv, float* __restrict__ enorm, int e) {
    int i = blockIdx.x * blockDim.x + threadIdx.x;
    if (i < e) enorm[i] = dinv[src[i]] * dinv[dst[i]];
}

// ---------------- WMMA fp32 GEMM: H[N,Cout] = X[N,Cin] @ W[Cin,Cout] ----------------
// block = 128 threads (4 waves); blockIdx.x = M-tile (16 rows), blockIdx.y*4+wave = N-tile.
// A tile staged to LDS with async-to-LDS DMA (ASYNCcnt), then fp32 WMMA over K.
template<int CIN, int COUT>
__global__ void k_gemm(const float* __restrict__ X, const float* __restrict__ W,
                       float* __restrict__ H) {
    __shared__ float sA[16 * CIN];   // sole LDS object -> LDS offset 0
    int tid = threadIdx.x;
    int mt = blockIdx.x;
    const float* Xrow = X + (size_t)mt * 16 * CIN;

    // async copy 16*CIN floats, 16B per lane per issue (addresses 16B aligned)
    for (int i = tid * 4; i < 16 * CIN; i += 128 * 4) {
        unsigned lds_off = (unsigned)(i * 4);
        const float* gp = Xrow + i;
        asm volatile("global_load_async_to_lds_b128 %0, %1, off"
                     :: "v"(lds_off), "v"(gp) : "memory");
    }
    asm volatile("s_wait_asynccnt 0x0" ::: "memory");
    __syncthreads();

    int wave = tid >> 5, lane = tid & 31;
    int ntile = blockIdx.y * 4 + wave;
    if (ntile * 16 >= COUT) return;   // wave-uniform: EXEC stays all-ones for WMMA

    int m  = lane & 15;               // A row within tile
    int kb = (lane >> 4) << 1;        // K slot: lanes16-31 hold K+2,K+3
    int col = ntile * 16 + (lane & 15);

    v8f acc = {0.f, 0.f, 0.f, 0.f, 0.f, 0.f, 0.f, 0.f};
#pragma unroll 4
    for (int k0 = 0; k0 < CIN; k0 += 4) {
        v2f a, b;
        a.x = sA[m * CIN + k0 + kb];
        a.y = sA[m * CIN + k0 + kb + 1];
        b.x = W[(size_t)(k0 + kb) * COUT + col];
        b.y = W[(size_t)(k0 + kb + 1) * COUT + col];
        acc = __builtin_amdgcn_wmma_f32_16x16x4_f32(
            false, a, false, b, (short)0, acc, false, false);
    }
    int rbase = (lane >> 4) * 8;      // C/D: VGPR r -> row r (lanes0-15) / r+8 (lanes16-31)
#pragma unroll
    for (int r = 0; r < 8; ++r)
        H[((size_t)mt * 16 + rbase + r) * COUT + col] = acc[r];
}

// ---------------- Agg init: self-loop + bias + time embedding ----------------
template<int COUT, bool HAS_TE>
__global__ void k_node_init(const float* __restrict__ H, const float* __restrict__ dinv2,
                            const float* __restrict__ bias, const float* __restrict__ te,
                            float* __restrict__ A) {
    long long idx = (long long)blockIdx.x * blockDim.x + threadIdx.x;
    if (idx >= (long long)NN * COUT) return;
    int c = (int)(idx % COUT);
    int row = (int)(idx / COUT);
    float v = H[idx] * dinv2[row] + bias[c];
    if (HAS_TE) v += te[c];
    A[idx] = v;
}

// ---------------- edge scatter: A[dst] += H[src] * enorm ----------------
template<int COUT>
__global__ void k_scatter(const float* __restrict__ H, const float* __restrict__ enorm,
                          const int* __restrict__ src, const int* __restrict__ dst,
                          float* __restrict__ A) {
    constexpr int C4 = COUT / 4;
    long long idx = (long long)blockIdx.x * blockDim.x + threadIdx.x;
    if (idx >= (long long)EE * C4) return;
    int q = (int)(idx % C4);
    int e = (int)(idx / C4);
    float nr = enorm[e];
    int s = src[e], d = dst[e];
    const float4 h = *reinterpret_cast<const float4*>(H + (size_t)s * COUT + q * 4);
    float* ap = A + (size_t)d * COUT + q * 4;
    atomAddF(ap + 0, h.x * nr);
    atomAddF(ap + 1, h.y * nr);
    atomAddF(ap + 2, h.z * nr);
    atomAddF(ap + 3, h.w * nr);
}

// ---------------- batchnorm stats (sum, sumsq per channel) ----------------
template<int COUT>
__global__ void k_stats(const float* __restrict__ A, float* __restrict__ sums,
                        float* __restrict__ sumsq, int prerelu) {
    int r0 = blockIdx.x * 128;
    int rend = min(r0 + 128, NN);
    for (int c = threadIdx.x; c < COUT; c += blockDim.x) {
        float s = 0.f, s2 = 0.f;
        for (int r = r0; r < rend; ++r) {
            float v = A[(size_t)r * COUT + c];
            if (prerelu) v = fmaxf(v, 0.f);
            s += v; s2 += v * v;
        }
        atomAddF(&sums[c], s);
        atomAddF(&sumsq[c], s2);
    }
}

// ---------------- batchnorm apply (+pre/post relu) ----------------
template<int COUT>
__global__ void k_norm(const float* __restrict__ A, const float* __restrict__ sums,
                       const float* __restrict__ sumsq, const float* __restrict__ g,
                       const float* __restrict__ be, float* __restrict__ Y,
                       int prerelu, int postrelu) {
    constexpr float invN = 1.f / NN;
    long long idx = (long long)blockIdx.x * blockDim.x + threadIdx.x;
    if (idx >= (long long)NN * COUT) return;
    int c = (int)(idx % COUT);
    float mean = sums[c] * invN;
    float var  = sumsq[c] * invN - mean * mean;
    float inv  = rsqrtf(var + BN_EPS);
    float v = A[idx];
    if (prerelu) v = fmaxf(v, 0.f);
    float y = g[c] * (v - mean) * inv + be[c];
    if (postrelu) y = fmaxf(y, 0.f);
    Y[idx] = y;
}

struct Common {
    const int* src; const int* dst;
    const float* dinv2; const float* enorm;
    float* sums; float* sumsq;
    hipStream_t stream;
};

template<int CIN, int COUT>
static void run_layer(const Common& cm, const float* X, const float* W, const float* bias,
                      const float* te, const float* g, const float* be,
                      float* H, float* A, float* Y, int prerelu, int postrelu) {
    dim3 gg(NN / 16, (COUT / 16 + 3) / 4);
    k_gemm<CIN, COUT><<<gg, 128, 0, cm.stream>>>(X, W, H);
    long long tot = (long long)NN * COUT;
    int nb1 = (int)((tot + 255) / 256);
    if (te)
        k_node_init<COUT, true><<<nb1, 256, 0, cm.stream>>>(H, cm.dinv2, bias, te, A);
    else
        k_node_init<COUT, false><<<nb1, 256, 0, cm.stream>>>(H, cm.dinv2, bias, te, A);
    long long stot = (long long)EE * (COUT / 4);
    k_scatter<COUT><<<(int)((stot + 255) / 256), 256, 0, cm.stream>>>(H, cm.enorm,
                                                                      cm.src, cm.dst, A);
    k_fill<<<4, 256, 0, cm.stream>>>(cm.sums, 1024, 0.f);
    k_stats<COUT><<<(NN + 127) / 128, 256, 0, cm.stream>>>(A, cm.sums, cm.sumsq, prerelu);
    k_norm<COUT><<<nb1, 256, 0, cm.stream>>>(A, cm.sums, cm.sumsq, g, be, Y,
                                             prerelu, postrelu);
}

extern "C" void kernel_launch(void* const* d_in, const int* in_sizes, int n_in,
                              void* d_out, int out_size, void* d_ws, size_t ws_size,
                              hipStream_t stream) {
    (void)in_sizes; (void)n_in; (void)out_size; (void)ws_size;
    const float* x     = (const float*)d_in[0];
    const int*   src   = (const int*)d_in[1];
    const int*   dst   = src + EE;
    const float* ts    = (const float*)d_in[2];
    const float* Wtm   = (const float*)d_in[3];
    const float* btm   = (const float*)d_in[4];
    const float* W_in  = (const float*)d_in[5];
    const float* b_in  = (const float*)d_in[6];
    const float* g_in  = (const float*)d_in[7];
    const float* be_in = (const float*)d_in[8];
    const float* Wt1   = (const float*)d_in[9];
    const float* bt1   = (const float*)d_in[10];
    const float* Wc1   = (const float*)d_in[11];
    const float* bc1   = (const float*)d_in[12];
    const float* g1    = (const float*)d_in[13];
    const float* be1   = (const float*)d_in[14];
    const float* Wt2   = (const float*)d_in[15];
    const float* bt2   = (const float*)d_in[16];
    const float* Wc2   = (const float*)d_in[17];
    const float* bc2   = (const float*)d_in[18];
    const float* g2    = (const float*)d_in[19];
    const float* be2   = (const float*)d_in[20];
    const float* W_out = (const float*)d_in[21];
    const float* b_out = (const float*)d_in[22];
    const float* g_out = (const float*)d_in[23];
    const float* be_out= (const float*)d_in[24];

    float* ws = (float*)d_ws;
    size_t nb = (size_t)NN * C9;
    float* B0    = ws;
    float* B1    = ws + nb;
    float* B2    = ws + 2 * nb;
    float* dinv  = ws + 3 * nb;
    float* dinv2 = dinv + NN;
    float* enorm = dinv2 + NN;
    float* te1   = enorm + EE;
    float* te2   = te1 + C9;
    float* sums  = te2 + C3;      // 1024-float region: [0,432) sums, [432,864) sumsq
    float* sumsq = sums + C9;

    // ---- preliminaries ----
    k_time<<<1, 512, 0, stream>>>(ts, Wtm, btm, Wt1, bt1, Wt2, bt2, te1, te2);
    k_fill<<<(NN + 255) / 256, 256, 0, stream>>>(dinv, NN, 1.f);          // deg = 1 (self loop)
    k_degcnt<<<(EE + 255) / 256, 256, 0, stream>>>(dst, dinv, EE);
    k_dinv<<<(NN + 255) / 256, 256, 0, stream>>>(dinv, dinv, dinv2, NN);  // in-place
    k_enorm<<<(EE + 255) / 256, 256, 0, stream>>>(src, dst, dinv, enorm, EE);

    Common cm{src, dst, dinv2, enorm, sums, sumsq, stream};

    // L0: input conv 48->144, relu(BN(.))
    run_layer<CC, C3>(cm, x,  W_in,  b_in,  nullptr, g_in,  be_in,  B1, B2, B0, 0, 1);
    // L1: block1 144->432, BN(relu(. + te1))
    run_layer<C3, C9>(cm, B0, Wc1,   bc1,   te1,     g1,    be1,    B1, B2, B1, 1, 0);
    // L2: block2 432->144, BN(relu(. + te2))
    run_layer<C9, C3>(cm, B1, Wc2,   bc2,   te2,     g2,    be2,    B0, B2, B0, 1, 0);
    // L3: output conv 144->48, relu(BN(.))
    run_layer<C3, CC>(cm, B0, W_out, b_out, nullptr, g_out, be_out, B1, B2,
                      (float*)d_out, 0, 1);
}